// Loss_Class_63385127354674
// MI455X (gfx1250) — compile-verified
//
#include <hip/hip_runtime.h>
#include <stdint.h>

// Problem constants (match reference: N=30, K=100, ALPHA=0.5 -> 1/(1-ALPHA)=2)
#define NROW    128            // rows per block == threads per block (4 waves)
#define NDELTA  56             // 2*(N-2)
#define NS      31             // N+1
#define KSTRIKE 100.0f

typedef __attribute__((ext_vector_type(2))) float v2f;
typedef __attribute__((ext_vector_type(8))) float v8f;

// ---------------- CDNA5 async global->LDS copies (ASYNCcnt path) -------------
__device__ __forceinline__ void async_load_b128(uint32_t lds_off, uint64_t gaddr) {
  asm volatile("global_load_async_to_lds_b128 %0, %1, off"
               :: "v"(lds_off), "v"(gaddr) : "memory");
}
__device__ __forceinline__ void async_load_b32(uint32_t lds_off, uint64_t gaddr) {
  asm volatile("global_load_async_to_lds_b32 %0, %1, off"
               :: "v"(lds_off), "v"(gaddr) : "memory");
}
__device__ __forceinline__ void wait_async_zero() {
#if __has_builtin(__builtin_amdgcn_s_wait_asynccnt)
  __builtin_amdgcn_s_wait_asynccnt(0);
#else
  asm volatile("s_wait_asynccnt 0" ::: "memory");
#endif
}

// Flat, coalesced tile copy: 16B per lane per iteration; b32 tail.
// Low 32 bits of a generic pointer to __shared__ are the wave-relative LDS
// offset on AMDGPU (flat LDS aperture truncates to addr[31:0]).
__device__ __forceinline__ void async_copy_tile(const float* __restrict__ gsrc,
                                                const float* lds_dst,
                                                int nbytes, int tid, int nthr) {
  uint64_t g = (uint64_t)(uintptr_t)gsrc;
  uint32_t l = (uint32_t)(uintptr_t)lds_dst;
  const int nb16 = nbytes & ~15;
  for (int off = tid * 16; off < nb16; off += nthr * 16)
    async_load_b128(l + (uint32_t)off, g + (uint64_t)off);
  const int rem_words = (nbytes - nb16) >> 2;
  if (tid < rem_words) {
    const int off = nb16 + tid * 4;
    async_load_b32(l + (uint32_t)off, g + (uint64_t)off);
  }
}

// ---------------- wave32 sum via WMMA (layout-agnostic ones-matvec) ----------
// D = A(16x4) x Ones(4x16): D[m][n] = rowsum(m). Sum of D VGPRs 0..7 in a lane
// = sum of rowsums for half the rows; shfl_xor(16) merges the two halves, so
// every lane ends with the sum of all 64 A slots (= 32 lanes x 2 values).
// Correct for ANY lane->(m,k) striping of A. Requires EXEC == all ones.
__device__ __forceinline__ float wave_sum32(float v) {
#if __has_builtin(__builtin_amdgcn_wmma_f32_16x16x4_f32)
  v2f a; a[0] = v;    a[1] = 0.0f;
  v2f b; b[0] = 1.0f; b[1] = 1.0f;
  v8f c = {};
  c = __builtin_amdgcn_wmma_f32_16x16x4_f32(false, a, false, b,
                                            (short)0, c, false, false);
  float s = c[0] + c[1] + c[2] + c[3] + c[4] + c[5] + c[6] + c[7];
  return s + __shfl_xor(s, 16, 32);
#else
  for (int m = 16; m >= 1; m >>= 1) v += __shfl_xor(v, m, 32);
  return v;
#endif
}

// ---------------- kernel 1: per-row loss + per-block partial sum -------------
__global__ __launch_bounds__(NROW) void loss_blocksum_kernel(
    const float* __restrict__ delta, const float* __restrict__ bx,
    const float* __restrict__ S1, const float* __restrict__ S2,
    const float* __restrict__ wptr, float* __restrict__ block_sums,
    long long nrows_total) {
  __shared__ float sDelta[NROW * NDELTA];  // 28672 B
  __shared__ float sS1[NROW * NS];         // 15872 B
  __shared__ float sS2[NROW * NS];         // 15872 B
  __shared__ float sBx[NROW * 2];          //  1024 B
  __shared__ float sWave[NROW / 32];

  const int tid = threadIdx.x;
  const long long row0 = (long long)blockIdx.x * NROW;
  const long long rem  = nrows_total - row0;
  const int vr = (rem < (long long)NROW) ? (int)rem : NROW;

  async_copy_tile(delta + row0 * NDELTA, sDelta, vr * NDELTA * 4, tid, NROW);
  async_copy_tile(S1 + row0 * NS,        sS1,    vr * NS * 4,     tid, NROW);
  async_copy_tile(S2 + row0 * NS,        sS2,    vr * NS * 4,     tid, NROW);
  async_copy_tile(bx + row0 * 2,         sBx,    vr * 2 * 4,      tid, NROW);
  wait_async_zero();
  __syncthreads();

  const float wv = wptr[0];
  const float* d  = &sDelta[tid * NDELTA];
  const float* s1 = &sS1[tid * NS];
  const float* s2 = &sS2[tid * NS];

  float sds = sBx[tid * 2] * (s1[1] - s1[0]) + sBx[tid * 2 + 1] * (s2[1] - s2[0]);
#pragma unroll
  for (int j = 0; j < NDELTA / 2; ++j)
    sds += d[2 * j]     * (s1[j + 2] - s1[j + 1])
        +  d[2 * j + 1] * (s2[j + 2] - s2[j + 1]);

  const float liab = fmaxf(s1[NS - 1] - KSTRIKE, 0.0f);
  float loss = wv + 2.0f * fmaxf(liab - sds - wv, 0.0f);
  if (tid >= vr) loss = 0.0f;              // tail rows contribute 0 (selects, no divergence at WMMA)

  const float wsum = wave_sum32(loss);
  if ((tid & 31) == 0) sWave[tid >> 5] = wsum;
  __syncthreads();
  if (tid == 0) {
    float t = 0.0f;
#pragma unroll
    for (int i = 0; i < NROW / 32; ++i) t += sWave[i];
    block_sums[blockIdx.x] = t;
  }
}

// ---------------- kernel 2: deterministic final reduction --------------------
__global__ __launch_bounds__(256) void final_reduce_kernel(
    const float* __restrict__ bsums, int n, float invB, float* __restrict__ out) {
  __shared__ float sWave[8];
  float p = 0.0f;
  for (int i = threadIdx.x; i < n; i += 256) p += bsums[i];
  for (int m = 16; m >= 1; m >>= 1) p += __shfl_xor(p, m, 32);
  if ((threadIdx.x & 31) == 0) sWave[threadIdx.x >> 5] = p;
  __syncthreads();
  if (threadIdx.x == 0) {
    float t = 0.0f;
#pragma unroll
    for (int i = 0; i < 8; ++i) t += sWave[i];
    out[0] = t * invB;
  }
}

// ---------------- host entry -------------------------------------------------
extern "C" void kernel_launch(void* const* d_in, const int* in_sizes, int n_in,
                              void* d_out, int out_size, void* d_ws, size_t ws_size,
                              hipStream_t stream) {
  const float* delta = (const float*)d_in[0];   // (B, 56)
  const float* bx    = (const float*)d_in[1];   // (B, 2)
  const float* S1    = (const float*)d_in[2];   // (B, 31)
  const float* S2    = (const float*)d_in[3];   // (B, 31)
  const float* w     = (const float*)d_in[4];   // scalar

  const long long B = (long long)in_sizes[0] / NDELTA;
  const int nblocks = (int)((B + NROW - 1) / NROW);
  float* bsums = (float*)d_ws;                   // nblocks floats (written, not accumulated)

  loss_blocksum_kernel<<<nblocks, NROW, 0, stream>>>(delta, bx, S1, S2, w, bsums, B);
  final_reduce_kernel<<<1, 256, 0, stream>>>(bsums, nblocks,
                                             (float)(1.0 / (double)B),
                                             (float*)d_out);
}